// Seq2Seq_30176440222472
// MI455X (gfx1250) — compile-verified
//
#include <hip/hip_runtime.h>

#define HID 8
#define T_IN 512
#define T_OUT 128
#define BATCH 65536

typedef __attribute__((ext_vector_type(2))) float v2f;
typedef __attribute__((ext_vector_type(8))) float v8f;

// Swap values between the two 16-lane halves of a wave32 (lane ^= 16).
// v_permlanex16_b32 with identity selectors = pure-VALU half swap (no DScnt).
__device__ __forceinline__ float half_swap(float v) {
#if __has_builtin(__builtin_amdgcn_permlanex16)
  return __int_as_float(__builtin_amdgcn_permlanex16(
      __float_as_int(v), __float_as_int(v), 0x76543210, 0xFEDCBA98, false, false));
#else
  return __int_as_float(__builtin_amdgcn_ds_swizzle(__float_as_int(v), 0x401f)); // SWAPX16
#endif
}

struct GruW {
  // A-operand K-tiles for V_WMMA_F32_16X16X4_F32 over augmented K = [h(8); x; 1; 0; 0]
  v2f aRZ[3];  // rows = gates 0..15 (r|z), pre-scaled by -log2(e)  -> sigmoid = rcp(1+exp2(d))
  v2f aNH[3];  // rows = gates 16..23 hidden part (+ b_hh_n), pre-scaled by -2*log2(e)
  v2f aNI;     // K-tile 2 only: n-gate input part (x*W_ih_n + b_ih_n), pre-scaled by -2*log2(e)
};

__device__ __forceinline__ void load_gru(GruW& W,
    const float* __restrict__ W_ih, const float* __restrict__ W_hh,
    const float* __restrict__ b_ih, const float* __restrict__ b_hh, int lane) {
  const int m  = lane & 15;
  const int hi = lane >> 4;                 // 0 = lanes 0-15, 1 = lanes 16-31
  const float sRZ = -1.44269504f;           // -log2(e)
  const float sN  = -2.88539008f;           // -2*log2(e)
  // f32 A layout: VGPR0 = K {k0 | k0+2} across halves, VGPR1 = +1
#pragma unroll
  for (int kt = 0; kt < 2; ++kt) {
    const int k = kt * 4 + hi * 2;
    W.aRZ[kt] = (v2f){ sRZ * W_hh[m * HID + k], sRZ * W_hh[m * HID + k + 1] };
    if (m < 8) {
      const int r = 16 + m;
      W.aNH[kt] = (v2f){ sN * W_hh[r * HID + k], sN * W_hh[r * HID + k + 1] };
    } else {
      W.aNH[kt] = (v2f){0.f, 0.f};
    }
  }
  // K-tile 2: global K = 8 (x), 9 (1), 10..11 (pad). Low lanes hold K=8,9.
  if (hi == 0) {
    W.aRZ[2] = (v2f){ sRZ * W_ih[m], sRZ * (b_ih[m] + b_hh[m]) };
    W.aNH[2] = (m < 8) ? (v2f){ 0.f, sN * b_hh[16 + m] } : (v2f){0.f, 0.f};
    W.aNI    = (m < 8) ? (v2f){ sN * W_ih[16 + m], sN * b_ih[16 + m] } : (v2f){0.f, 0.f};
  } else {
    W.aRZ[2] = (v2f){0.f, 0.f};
    W.aNH[2] = (v2f){0.f, 0.f};
    W.aNI    = (v2f){0.f, 0.f};
  }
}

__device__ __forceinline__ void gru_step(float h[8], float xv, const GruW& W,
                                         bool lo, float b2c) {
  // B operand: [h; x; 1] as 12x16 (K x batch), three 4x16 K-tiles.
  // B layout: VGPR0 = K rows {0 | 2} across halves, VGPR1 = {1 | 3}.
  const float s2 = half_swap(h[2]);
  const float s3 = half_swap(h[3]);
  const float s6 = half_swap(h[6]);
  const float s7 = half_swap(h[7]);
  const v2f b0 = (v2f){ lo ? h[0] : s2, lo ? h[1] : s3 };   // K 0..3
  const v2f b1 = (v2f){ lo ? h[4] : s6, lo ? h[5] : s7 };   // K 4..7
  const v2f b2 = (v2f){ lo ? xv : 0.f, b2c };               // K 8..11 = x,1,0,0
  const v8f zero = {};

  // d0 = -log2e * (W_hh_rz@h + W_ih_rz*x + b_rz)   (low: r, high: z)
  v8f d0 = __builtin_amdgcn_wmma_f32_16x16x4_f32(false, W.aRZ[0], false, b0,
                                                 (short)0, zero, false, false);
  d0     = __builtin_amdgcn_wmma_f32_16x16x4_f32(false, W.aRZ[1], false, b1,
                                                 (short)0, d0, false, false);
  d0     = __builtin_amdgcn_wmma_f32_16x16x4_f32(false, W.aRZ[2], false, b2,
                                                 (short)0, d0, false, false);
  // d1 = -2log2e * (W_hh_n@h + b_hh_n)   (h_n, low lanes)
  v8f d1 = __builtin_amdgcn_wmma_f32_16x16x4_f32(false, W.aNH[0], false, b0,
                                                 (short)0, zero, false, false);
  d1     = __builtin_amdgcn_wmma_f32_16x16x4_f32(false, W.aNH[1], false, b1,
                                                 (short)0, d1, false, false);
  d1     = __builtin_amdgcn_wmma_f32_16x16x4_f32(false, W.aNH[2], false, b2,
                                                 (short)0, d1, false, false);
  // d2 = -2log2e * (W_ih_n*x + b_ih_n)   (i_n, low lanes)
  const v8f d2 = __builtin_amdgcn_wmma_f32_16x16x4_f32(false, W.aNI, false, b2,
                                                       (short)0, zero, false, false);

  float g[8];
#pragma unroll
  for (int j = 0; j < 8; ++j) {
    // sigmoid(pre) = rcp(1 + exp2(-log2e*pre)); scale pre-folded into weights
    g[j] = __builtin_amdgcn_rcpf(1.0f + __builtin_amdgcn_exp2f(d0[j]));
  }
#pragma unroll
  for (int j = 0; j < 8; ++j) {
    const float zg = half_swap(g[j]);                      // z gate into low lanes
    // tanh(npre) = 2*rcp(1+exp2(-2log2e*npre)) - 1 ; scales pre-folded
    const float q  = __builtin_amdgcn_rcpf(
        1.0f + __builtin_amdgcn_exp2f(fmaf(g[j], d1[j], d2[j])));
    const float nv = fmaf(2.0f, q, -1.0f);
    h[j] = fmaf(zg, h[j] - nv, nv);                        // (1-z)*n + z*h
  }
}

__global__ __launch_bounds__(256) void seq2seq_gru_wmma(
    const float* __restrict__ x,
    const float* __restrict__ eWih, const float* __restrict__ eWhh,
    const float* __restrict__ eBih, const float* __restrict__ eBhh,
    const float* __restrict__ dWih, const float* __restrict__ dWhh,
    const float* __restrict__ dBih, const float* __restrict__ dBhh,
    const float* __restrict__ fcW,  const float* __restrict__ fcB,
    float* __restrict__ out) {
  const int lane = threadIdx.x & 31;
  const int wave = (int)((blockIdx.x * blockDim.x + threadIdx.x) >> 5);
  const int base = wave * 16;            // 16 batch elements per wave
  const int bcol = base + (lane & 15);   // batch index, broadcast to both halves
  const bool lo  = (lane < 16);
  const float b2c = lo ? 1.0f : 0.0f;    // constant "bias row" of the B operand

  GruW We;
  load_gru(We, eWih, eWhh, eBih, eBhh, lane);

  float h[8];
#pragma unroll
  for (int j = 0; j < 8; ++j) h[j] = 0.f;

  // ---- encoder: 512 serial steps over streamed x ----
  for (int t = 0; t < T_IN; ++t) {
    const float xv = x[t * BATCH + bcol];
    gru_step(h, xv, We, lo, b2c);
  }

  // ---- decoder: 128 serial steps, feed back fc(h) ----
  GruW Wd;
  load_gru(Wd, dWih, dWhh, dBih, dBhh, lane);
  float fw[8];
#pragma unroll
  for (int j = 0; j < 8; ++j) fw[j] = fcW[j];
  const float fb = fcB[0];

  float inp = x[(T_IN - 1) * BATCH + bcol];   // x[-1]
  for (int t = 0; t < T_OUT; ++t) {
    gru_step(h, inp, Wd, lo, b2c);
    float o = fb;
#pragma unroll
    for (int j = 0; j < 8; ++j) o = fmaf(h[j], fw[j], o);
    if (lane < 16) out[t * BATCH + base + lane] = o;
    const float obc = half_swap(o);           // broadcast to high half
    inp = lo ? o : obc;
  }
}

extern "C" void kernel_launch(void* const* d_in, const int* in_sizes, int n_in,
                              void* d_out, int out_size, void* d_ws, size_t ws_size,
                              hipStream_t stream) {
  (void)in_sizes; (void)n_in; (void)out_size; (void)d_ws; (void)ws_size;
  const float* x    = (const float*)d_in[0];
  const float* eWih = (const float*)d_in[1];
  const float* eWhh = (const float*)d_in[2];
  const float* eBih = (const float*)d_in[3];
  const float* eBhh = (const float*)d_in[4];
  const float* dWih = (const float*)d_in[5];
  const float* dWhh = (const float*)d_in[6];
  const float* dBih = (const float*)d_in[7];
  const float* dBhh = (const float*)d_in[8];
  const float* fcW  = (const float*)d_in[9];
  const float* fcB  = (const float*)d_in[10];
  float* out = (float*)d_out;

  // 16 batch / wave, 8 waves / block of 256 -> 128 batch / block
  const int blocks = BATCH / 128;  // 512
  seq2seq_gru_wmma<<<blocks, 256, 0, stream>>>(
      x, eWih, eWhh, eBih, eBhh, dWih, dWhh, dBih, dBhh, fcW, fcB, out);
}